// Model_87943750353492
// MI455X (gfx1250) — compile-verified
//
#include <hip/hip_runtime.h>
#include <stdint.h>

// Problem constants (from reference)
#define T_STEPS 1000
#define NBATCH  32
#define R       1024        // CRF states
#define NB1     5           // scores per state
#define CROW    (R * NB1)   // 5120 floats per (t, n)
#define ROWB    (CROW * 4)  // 20480 bytes per (t, n)

#define LOG2E_F 1.4426950408889634f
#define LN2_F   0.6931471805599453f

// Hardware transcendentals (base-2 on AMDGCN): v_exp_f32 / v_log_f32
#define EXP2F(x) __builtin_amdgcn_exp2f(x)
#define LOG2F(x) __builtin_amdgcn_logf(x)

// ---- CDNA5 async global->LDS helpers (inline asm; portable across toolchains) ----

// Byte offset of a __shared__ object within LDS (generic -> AS(3) addrspacecast truncates).
__device__ __forceinline__ uint32_t lds_byte_off(const void* p) {
    return (uint32_t)(uintptr_t)(__attribute__((address_space(3))) const void*)p;
}

__device__ __forceinline__ void async_copy_b128(uint32_t ldsByte, uint64_t sbase, uint32_t voff) {
    asm volatile("global_load_async_to_lds_b128 %0, %1, %2 th:TH_LOAD_NT"
                 :: "v"(ldsByte), "v"(voff), "s"(sbase) : "memory");
}
__device__ __forceinline__ void async_copy_b32(uint32_t ldsByte, uint64_t sbase, uint32_t voff) {
    asm volatile("global_load_async_to_lds_b32 %0, %1, %2 th:TH_LOAD_NT"
                 :: "v"(ldsByte), "v"(voff), "s"(sbase) : "memory");
}
__device__ __forceinline__ void wait_async_le2() {
    asm volatile("s_wait_asynccnt 2" ::: "memory");
}
__device__ __forceinline__ void wait_async_le0() {
    asm volatile("s_wait_asynccnt 0" ::: "memory");
}

__global__ __launch_bounds__(1024, 1)
void crf_logz_kernel(const float* __restrict__ scores, float* __restrict__ out) {
    // Triple-buffered score tile (60 KB) + double-buffered alpha (8 KB) = 68 KB LDS.
    __shared__ __align__(16) float Mbuf[3][CROW];
    __shared__ __align__(16) float alpha[2][R];

    const int s = threadIdx.x;     // state id, 0..1023
    const int n = blockIdx.x;      // batch id, 0..31
    const int q = s >> 2;          // shared predecessor base

    alpha[0][s] = 0.0f;            // log2-domain alpha_0 (== 0 in any base)

    const uint64_t base = (uint64_t)(uintptr_t)scores;
    const uint32_t mb0 = lds_byte_off(&Mbuf[0][0]);
    const uint32_t mb1 = lds_byte_off(&Mbuf[1][0]);
    const uint32_t mb2 = lds_byte_off(&Mbuf[2][0]);
    (void)mb2;

    const uint32_t off128 = (uint32_t)s * 16u;            // bytes [0, 16K)
    const uint32_t off32  = 16384u + (uint32_t)s * 4u;    // bytes [16K, 20K)

    // Every thread issues exactly 2 async instructions per tile -> uniform waits.
#define ISSUE_TILE(tt, LBASE)                                                   \
    do {                                                                        \
        const uint64_t g_ = base + ((uint64_t)(tt) * NBATCH + (uint64_t)n) * (uint64_t)ROWB; \
        async_copy_b128((LBASE) + off128, g_, off128);                          \
        async_copy_b32 ((LBASE) + off32,  g_, off32);                           \
    } while (0)

    // ---- Prologue: preload t = 0 and t = 1 ----
    ISSUE_TILE(0, mb0);
    ISSUE_TILE(1, mb1);
    wait_async_le2();   // t=0 tile landed (t=1 may still be in flight)
    __syncthreads();

    int p = 0;  // alpha read buffer

    // One pipeline stage: prefetch t+2, compute t, wait for t+1, barrier.
#define STEP(tt, CURB, NXT2B)                                                   \
    do {                                                                        \
        const bool pre_ = ((tt) + 2 < T_STEPS);                                 \
        if (pre_) ISSUE_TILE((tt) + 2, (NXT2B));                                \
        const float* __restrict__ M_  = &Mbuf[(CURB)][s * NB1];                 \
        const float* __restrict__ aP_ = &alpha[p][0];                           \
        const float a0_ = fmaf(M_[0], LOG2E_F, aP_[s]);                         \
        const float a1_ = fmaf(M_[1], LOG2E_F, aP_[q]);                         \
        const float a2_ = fmaf(M_[2], LOG2E_F, aP_[q + 256]);                   \
        const float a3_ = fmaf(M_[3], LOG2E_F, aP_[q + 512]);                   \
        const float a4_ = fmaf(M_[4], LOG2E_F, aP_[q + 768]);                   \
        const float mx_ = fmaxf(fmaxf(fmaxf(a0_, a1_), fmaxf(a2_, a3_)), a4_); \
        const float sm_ = EXP2F(a0_ - mx_) + EXP2F(a1_ - mx_)                   \
                        + EXP2F(a2_ - mx_) + EXP2F(a3_ - mx_)                   \
                        + EXP2F(a4_ - mx_);                                     \
        alpha[p ^ 1][s] = mx_ + LOG2F(sm_);                                     \
        p ^= 1;                                                                 \
        if (pre_) wait_async_le2(); else wait_async_le0();                      \
        __syncthreads();                                                        \
    } while (0)

#pragma unroll 1
    for (int t = 0; t < T_STEPS; t += 3) {
        STEP(t, 0, 2);
        if (t + 1 < T_STEPS) STEP(t + 1, 1, 0);
        if (t + 2 < T_STEPS) STEP(t + 2, 2, 1);
    }
#undef STEP
#undef ISSUE_TILE

    // ---- Final logsumexp over the 1024 states (log2 domain; reuse Mbuf[0]) ----
    float* red = &Mbuf[0][0];
    red[s] = alpha[p][s];
    __syncthreads();
    for (int off = 512; off > 0; off >>= 1) {
        if (s < off) red[s] = fmaxf(red[s], red[s + off]);
        __syncthreads();
    }
    const float mx = red[0];
    __syncthreads();
    red[s] = EXP2F(alpha[p][s] - mx);
    __syncthreads();
    for (int off = 512; off > 0; off >>= 1) {
        if (s < off) red[s] = red[s] + red[s + off];
        __syncthreads();
    }
    if (s == 0) out[n] = LN2_F * (mx + LOG2F(red[0]));
}

extern "C" void kernel_launch(void* const* d_in, const int* in_sizes, int n_in,
                              void* d_out, int out_size, void* d_ws, size_t ws_size,
                              hipStream_t stream) {
    (void)in_sizes; (void)n_in; (void)d_ws; (void)ws_size; (void)out_size;
    const float* scores = (const float*)d_in[0];   // [T=1000, N=32, C=5120] fp32
    float* out = (float*)d_out;                    // [N=32] fp32
    crf_logz_kernel<<<dim3(NBATCH), dim3(R), 0, stream>>>(scores, out);
}